// CenterLoss_40965398069570
// MI455X (gfx1250) — compile-verified
//
#include <hip/hip_runtime.h>

#define NB_CLASS    100000
#define DIM         128
#define BATCH       16384
#define LOSS_WEIGHT 0.01f
#define ALPHA       0.05f
#define NTILES      (BATCH / 16)   // 1024 WMMA tiles of 16 samples

typedef __attribute__((ext_vector_type(2))) float v2f;
typedef __attribute__((ext_vector_type(4))) float v4f;
typedef __attribute__((ext_vector_type(8))) float v8f;

// ---------------------------------------------------------------- kernel 1
__global__ void zero_counts_kernel(unsigned* counts) {
    int i = blockIdx.x * blockDim.x + threadIdx.x;
    if (i < NB_CLASS) counts[i] = 0u;
}

// ---------------------------------------------------------------- kernel 2
// Zero only the accumulation rows that will be touched (duplicates benign).
// grid*block == BATCH*DIM exactly (8192 x 256).
__global__ void zero_touched_kernel(const int* __restrict__ y,
                                    float* __restrict__ outc) {
    int idx = blockIdx.x * blockDim.x + threadIdx.x;
    int i = idx >> 7;          // sample
    int d = idx & 127;         // dim
    int c = y[i];
    outc[(size_t)c * DIM + d] = 0.0f;
}

// ---------------------------------------------------------------- kernel 3
// One 128-thread block per sample: diff = centers[y[i]] - x[i],
// scatter-add into output accumulation rows via L2 fp32 atomics.
__global__ void scatter_kernel(const float* __restrict__ x,
                               const int* __restrict__ y,
                               const float* __restrict__ centers,
                               float* __restrict__ outc,
                               unsigned* __restrict__ counts) {
    int i = blockIdx.x;
    int d = threadIdx.x;
    int c = y[i];
    float diff = centers[(size_t)c * DIM + d] - x[(size_t)i * DIM + d];
    atomicAdd(&outc[(size_t)c * DIM + d], diff);
    if (d == 0) atomicAdd(&counts[c], 1u);
}

// ---------------------------------------------------------------- kernel 4
// Loss via fp32 WMMA: per 16-row tile, G = diff * diff^T accumulated with
// V_WMMA_F32_16X16X4_F32 over K=128; diag(G) = per-row sum of squares.
// Exactly one wave32 per block; no divergence before the WMMAs.
__global__ void __launch_bounds__(32)
wmma_loss_kernel(const float* __restrict__ x,
                 const int* __restrict__ y,
                 const float* __restrict__ centers,
                 float* __restrict__ partial) {
    __shared__ float diff_s[16 * 132];   // stride 132 floats: conflict-free
    const int lane = threadIdx.x;
    const int r0   = blockIdx.x * 16;

    // Stage diff tile (16 x 128) into LDS with coalesced 16B loads.
    #pragma unroll 4
    for (int r = 0; r < 16; ++r) {
        int row = r0 + r;
        int c = y[row];
        v4f xv = *(const v4f*)(x + (size_t)row * DIM + lane * 4);
        v4f cv = *(const v4f*)(centers + (size_t)c * DIM + lane * 4);
        v4f dv = cv - xv;
        *(v4f*)(&diff_s[r * 132 + lane * 4]) = dv;
    }
    __syncthreads();

    // fp32 A-matrix 16x4 layout: lanes 0-15 hold K=0,1; lanes 16-31 hold K=2,3.
    // For G = A*A^T the fp32 B(4x16) layout is identical -> reuse same regs.
    const int mrow = lane & 15;
    const int koff = (lane >> 4) * 2;
    const float* rp = &diff_s[mrow * 132 + koff];

    v8f acc = {};
    #pragma unroll
    for (int k0 = 0; k0 < DIM; k0 += 4) {
        v2f a;
        a.x = rp[k0];
        a.y = rp[k0 + 1];
        acc = __builtin_amdgcn_wmma_f32_16x16x4_f32(
            /*neg_a=*/false, a, /*neg_b=*/false, a,
            /*c_mod=*/(short)0, acc, /*reuse_a=*/false, /*reuse_b=*/false);
    }

    // Extract diagonal (m,m) from the 16x16 f32 C layout:
    //  lanes 0-7  hold (m=lane)   at acc[lane]
    //  lanes 24-31 hold (m=lane-16) at acc[lane-24]
    int s = (lane < 8) ? lane : ((lane >= 24) ? (lane - 24) : -1);
    float dg = 0.0f;
    dg = (s == 0) ? acc[0] : dg;
    dg = (s == 1) ? acc[1] : dg;
    dg = (s == 2) ? acc[2] : dg;
    dg = (s == 3) ? acc[3] : dg;
    dg = (s == 4) ? acc[4] : dg;
    dg = (s == 5) ? acc[5] : dg;
    dg = (s == 6) ? acc[6] : dg;
    dg = (s == 7) ? acc[7] : dg;

    // wave32 reduction
    #pragma unroll
    for (int off = 16; off >= 1; off >>= 1)
        dg += __shfl_xor(dg, off, 32);

    if (lane == 0) partial[blockIdx.x] = dg;
}

// ---------------------------------------------------------------- kernel 5
// Streaming finalize: out = centers - ALPHA*upd/(count+1); untouched rows copy.
// grid*block == NB_CLASS*DIM exactly (50000 x 256).
__global__ void finalize_kernel(const float* __restrict__ centers,
                                const unsigned* __restrict__ counts,
                                float* __restrict__ outc) {
    size_t idx = (size_t)blockIdx.x * blockDim.x + threadIdx.x;
    int c = (int)(idx >> 7);
    unsigned cnt = counts[c];
    float cen = centers[idx];
    float o = cen;
    if (cnt != 0u) {
        float upd = outc[idx];
        o = cen - ALPHA * upd / (float)(cnt + 1u);
    }
    outc[idx] = o;
}

// ---------------------------------------------------------------- kernel 6
// Deterministic fixed-order reduction of tile partials -> d_out[0].
__global__ void loss_reduce_kernel(const float* __restrict__ partial,
                                   float* __restrict__ out_loss) {
    __shared__ float sm[256];
    int t = threadIdx.x;
    float s = 0.0f;
    for (int i = t; i < NTILES; i += 256) s += partial[i];
    sm[t] = s;
    __syncthreads();
    #pragma unroll
    for (int o = 128; o >= 1; o >>= 1) {
        if (t < o) sm[t] += sm[t + o];
        __syncthreads();
    }
    if (t == 0) out_loss[0] = LOSS_WEIGHT * sm[0] / (float)BATCH;
}

// ----------------------------------------------------------------
extern "C" void kernel_launch(void* const* d_in, const int* in_sizes, int n_in,
                              void* d_out, int out_size, void* d_ws, size_t ws_size,
                              hipStream_t stream) {
    const float* x       = (const float*)d_in[0];
    const int*   y       = (const int*)d_in[1];
    const float* centers = (const float*)d_in[2];

    float* out  = (float*)d_out;   // out[0] = loss, out[1..] = new_centers
    float* outc = out + 1;

    unsigned* counts  = (unsigned*)d_ws;                                   // 400000 B
    float*    partial = (float*)((char*)d_ws + NB_CLASS * sizeof(unsigned)); // 4096 B

    zero_counts_kernel<<<(NB_CLASS + 255) / 256, 256, 0, stream>>>(counts);
    zero_touched_kernel<<<(BATCH * DIM) / 256, 256, 0, stream>>>(y, outc);
    scatter_kernel<<<BATCH, DIM, 0, stream>>>(x, y, centers, outc, counts);
    wmma_loss_kernel<<<NTILES, 32, 0, stream>>>(x, y, centers, partial);
    finalize_kernel<<<(NB_CLASS * DIM) / 256, 256, 0, stream>>>(centers, counts, outc);
    loss_reduce_kernel<<<1, 256, 0, stream>>>(partial, out);
}